// StructuredTransformerEncoder_2542620639820
// MI455X (gfx1250) — compile-verified
//
#include <hip/hip_runtime.h>

#define DM 128
#define FFDIM 512
#define LAYERS 6
#define EPS 1e-5f

typedef float v2f __attribute__((ext_vector_type(2)));
typedef float v8f __attribute__((ext_vector_type(8)));
typedef unsigned int v4u __attribute__((ext_vector_type(4)));
typedef int v4i __attribute__((ext_vector_type(4)));
typedef int v8i __attribute__((ext_vector_type(8)));

#if defined(__AMDGCN__) && __has_builtin(__builtin_amdgcn_tensor_load_to_lds)
#define HAVE_TDM 1
#else
#define HAVE_TDM 0
#endif

// LDS strides (floats) chosen for bank-conflict-free WMMA fragment reads:
//  X tile stride 132: A readers (16 lanes, stride 132) hit banks {0,4,..,60}
//  W tile stride 136: B readers' two half-waves hit disjoint 16-bank ranges
#define XS_STRIDE 132
#define WS_STRIDE 136

__device__ __forceinline__ unsigned lds_off(const void* p) {
    // low 32 bits of a generic LDS address are the wave-relative LDS byte offset
    return (unsigned)(size_t)p;
}

#if HAVE_TDM
// Issue a TDM 2D tile load: tile_d0 x tile_d1 elements (f32), row stride
// `stride_el`, LDS padding pad_amt_code dwords every 128 dwords (interval code 6).
// This toolchain's builtin takes 6 args: (g0, g1, g2, g3, extra_g, cpol).
__device__ __forceinline__ void tdm_load_2d(unsigned lds_byte,
                                            const float* gptr,
                                            unsigned tile_d0, unsigned tile_d1,
                                            unsigned tensor_d1,
                                            unsigned long long stride_el,
                                            unsigned pad_amt_code) {
    unsigned long long ga = (unsigned long long)(size_t)gptr;
    unsigned td0 = tile_d0;                    // tensor_dim0 == tile_dim0 (rows full)
    v4u g0;
    g0.x = 1u;                                 // count=1, is_restore=0, gather off
    g0.y = lds_byte;                           // lds_addr [63:32]
    g0.z = (unsigned)(ga & 0xFFFFFFFFu);       // global_addr [95:64]
    g0.w = (unsigned)((ga >> 32) & 0x01FFFFFFu) | (2u << 30);  // addr hi | type=2
    v8i g1;
    g1[0] = (int)((2u << 16)                   // data_size = 4B
                | (1u << 20)                   // pad_enable
                | (6u << 22)                   // pad_interval = 128 dwords
                | (pad_amt_code << 25));       // pad_amount code (dwords-1)
    g1[1] = (int)((td0 & 0xFFFFu) << 16);                                  // tensor_dim0 lo
    g1[2] = (int)(((td0 >> 16) & 0xFFFFu) | ((tensor_d1 & 0xFFFFu) << 16));// td0 hi | td1 lo
    g1[3] = (int)(((tensor_d1 >> 16) & 0xFFFFu) | ((tile_d0 & 0xFFFFu) << 16)); // td1 hi | tile0
    g1[4] = (int)(tile_d1 & 0xFFFFu);          // tile_dim1 | tile_dim2=0
    g1[5] = (int)(stride_el & 0xFFFFFFFFull);  // tensor_dim0_stride lo
    g1[6] = (int)((stride_el >> 32) & 0xFFFFull); // stride hi | dim1_stride lo = 0
    g1[7] = 0;
    v4i gz4 = {0, 0, 0, 0};                    // groups 2/3 unused (2D tile)
    v8i gz8 = {0, 0, 0, 0, 0, 0, 0, 0};        // extra group (unused, zero)
    __builtin_amdgcn_tensor_load_to_lds(g0, g1, gz4, gz4, gz8, 0);
}
#endif

// ---------------------------------------------------------------------------
// Embedding: x = node_features @ emb_W + emb_b   (K = 20, too small for WMMA)
// ---------------------------------------------------------------------------
__global__ void embed_kernel(const float* __restrict__ nf, const float* __restrict__ W,
                             const float* __restrict__ b, float* __restrict__ x, int n) {
    int i = blockIdx.x * blockDim.x + threadIdx.x;
    if (i >= n * DM) return;
    int row = i / DM, col = i - row * DM;
    const float* nr = nf + (size_t)row * 20;
    float acc = b[col];
#pragma unroll
    for (int j = 0; j < 20; ++j) acc = fmaf(nr[j], W[j * DM + col], acc);
    x[i] = acc;
}

// ---------------------------------------------------------------------------
// WMMA f32 GEMM: Y[nrows, ncols] = X[nrows, K] @ W[K, ncols] (+bias) (+ReLU)
// block = 256 threads (8 waves); block owns 16 rows x 128 cols; wave w owns
// cols blockIdx.y*128 + w*16. K processed in chunks of 128: TDM stages the
// X tile (16x128) and W tile (128x128) into padded LDS, then the inner loop
// is pure ds_load + v_wmma_f32_16x16x4_f32 (exact f32 math).
// ---------------------------------------------------------------------------
__global__ void gemm_wmma_f32(const float* __restrict__ X, const float* __restrict__ W,
                              const float* __restrict__ bias, float* __restrict__ Y,
                              int nrows, int K, int ncols, int relu) {
    __shared__ float xs[16 * XS_STRIDE];       //  8.4 KB
    __shared__ float ws[128 * WS_STRIDE];      // 69.6 KB
    int row0 = blockIdx.x * 16;
    int tid = threadIdx.x;
    int lane = tid & 31;
    int wv   = tid >> 5;
    int colg = blockIdx.y * 128;
    int mn   = lane & 15;          // A row / B,C,D column within 16x16 tile
    int kk   = (lane >> 4) << 1;   // k sub-offset for A/B fragments

    v8f acc = {0.f, 0.f, 0.f, 0.f, 0.f, 0.f, 0.f, 0.f};
    const float* xrow = xs + mn * XS_STRIDE;
    const float* wcol = ws + wv * 16 + mn;

    for (int k0 = 0; k0 < K; k0 += 128) {
#if HAVE_TDM
        if (wv == 0) {
            tdm_load_2d(lds_off(xs), X + (size_t)row0 * K + k0,
                        128u, 16u, (unsigned)(nrows - row0),
                        (unsigned long long)K, 3u /* pad 4 dw -> stride 132 */);
            tdm_load_2d(lds_off(ws), W + (size_t)k0 * ncols + colg,
                        128u, 128u, 128u,
                        (unsigned long long)ncols, 7u /* pad 8 dw -> stride 136 */);
#if __has_builtin(__builtin_amdgcn_s_wait_tensorcnt)
            __builtin_amdgcn_s_wait_tensorcnt(0);
#else
            asm volatile("s_wait_tensorcnt 0x0" ::: "memory");
#endif
        }
        __syncthreads();
#else
        for (int i = tid; i < 16 * 128; i += 256) {
            int r = i >> 7, c = i & 127;
            int rr = row0 + r; if (rr >= nrows) rr = nrows - 1;
            xs[r * XS_STRIDE + c] = X[(size_t)rr * K + k0 + c];
        }
        for (int i = tid; i < 128 * 128; i += 256) {
            int r = i >> 7, c = i & 127;
            ws[r * WS_STRIDE + c] = W[(size_t)(k0 + r) * ncols + colg + c];
        }
        __syncthreads();
#endif
        // 32 WMMAs per chunk per wave, all operands from LDS
#pragma unroll 8
        for (int kc = 0; kc < 128; kc += 4) {
            v2f a, b;
            a.x = xrow[kc + kk];
            a.y = xrow[kc + kk + 1];
            const float* wp = wcol + (kc + kk) * WS_STRIDE;
            b.x = wp[0];
            b.y = wp[WS_STRIDE];
            acc = __builtin_amdgcn_wmma_f32_16x16x4_f32(false, a, false, b,
                                                        (short)0, acc, false, false);
        }
        __syncthreads();   // protect LDS tiles before next chunk's DMA
    }

    int half = lane >> 4;
    int col = colg + wv * 16 + mn;
    float bb = bias ? bias[col] : 0.f;
#pragma unroll
    for (int r = 0; r < 8; ++r) {
        int row = row0 + r + 8 * half;        // C/D layout: VGPR r -> M = r + 8*half
        if (row < nrows) {
            float vv = acc[r] + bb;
            if (relu) vv = fmaxf(vv, 0.f);
            Y[(size_t)row * ncols + col] = vv;
        }
    }
}

// ---------------------------------------------------------------------------
// Fused edge kernel: edge-MLP + per-head attention logits + softmax over the
// 8 heads + weighted-v scatter-add. One wave per edge, 4 channels per lane
// (head h = lane/4). Edge-MLP weights staged in LDS.
// ---------------------------------------------------------------------------
__global__ void edge_attn_kernel(const float* __restrict__ q, const float* __restrict__ k,
                                 const float* __restrict__ v,
                                 const float* __restrict__ eattr,
                                 const int* __restrict__ srcIdx, const int* __restrict__ dstIdx,
                                 const float* __restrict__ eW1, const float* __restrict__ eb1,
                                 const float* __restrict__ eW2, const float* __restrict__ eb2,
                                 float* __restrict__ agg, int nedges) {
    __shared__ float w1[48];
    __shared__ float b1[16];
    __shared__ float w2[16 * DM];
    __shared__ float b2[DM];
    int tid = threadIdx.x;
    if (tid < 48) w1[tid] = eW1[tid];
    if (tid < 16) b1[tid] = eb1[tid];
    for (int i = tid; i < 16 * DM; i += blockDim.x) w2[i] = eW2[i];
    if (tid < DM) b2[tid] = eb2[tid];
    __syncthreads();

    const int lane = tid & 31;
    const int c0 = lane * 4;               // 4 contiguous channels per lane
    const float scale = 0.25f;             // 1/sqrt(D=16)
    int waveId = (blockIdx.x * blockDim.x + tid) >> 5;
    int nwaves = (gridDim.x * blockDim.x) >> 5;

    for (int e = waveId; e < nedges; e += nwaves) {
        int s = srcIdx[e];
        int d = dstIdx[e];
        const float4 qd = *(const float4*)(q + (size_t)d * DM + c0);
        const float4 ks = *(const float4*)(k + (size_t)s * DM + c0);
        const float4 vs = *(const float4*)(v + (size_t)s * DM + c0);

        // prefetch the next edge's rows (global_prefetch_b8)
        int e2 = e + nwaves;
        if (e2 < nedges) {
            int s2 = srcIdx[e2], d2 = dstIdx[e2];
            __builtin_prefetch(q + (size_t)d2 * DM + c0, 0, 1);
            __builtin_prefetch(k + (size_t)s2 * DM + c0, 0, 1);
            __builtin_prefetch(v + (size_t)s2 * DM + c0, 0, 1);
        }

        // edge MLP: hidden(16) = relu(attr(3) @ eW1 + eb1); ee = hidden @ eW2 + eb2
        float a0 = eattr[e * 3 + 0];
        float a1 = eattr[e * 3 + 1];
        float a2 = eattr[e * 3 + 2];
        float hb[16];
#pragma unroll
        for (int j = 0; j < 16; ++j) {
            float t = fmaf(a0, w1[j], fmaf(a1, w1[16 + j], fmaf(a2, w1[32 + j], b1[j])));
            hb[j] = fmaxf(t, 0.f);
        }
        float ee0 = b2[c0 + 0], ee1 = b2[c0 + 1], ee2 = b2[c0 + 2], ee3 = b2[c0 + 3];
#pragma unroll
        for (int j = 0; j < 16; ++j) {
            const float* wr = w2 + j * DM + c0;
            ee0 = fmaf(hb[j], wr[0], ee0);
            ee1 = fmaf(hb[j], wr[1], ee1);
            ee2 = fmaf(hb[j], wr[2], ee2);
            ee3 = fmaf(hb[j], wr[3], ee3);
        }

        // per-head logit: reduce q . (k + ee) over the head's 4 lanes
        float part = qd.x * (ks.x + ee0) + qd.y * (ks.y + ee1)
                   + qd.z * (ks.z + ee2) + qd.w * (ks.w + ee3);
        part += __shfl_xor(part, 1, 32);
        part += __shfl_xor(part, 2, 32);
        float logit = part * scale;

        // softmax over the 8 heads (groups of 4 lanes uniform; butterfly 4,8,16)
        float mx = logit;
        mx = fmaxf(mx, __shfl_xor(mx, 4, 32));
        mx = fmaxf(mx, __shfl_xor(mx, 8, 32));
        mx = fmaxf(mx, __shfl_xor(mx, 16, 32));
        float ex = __expf(logit - mx);
        float sm = ex;
        sm += __shfl_xor(sm, 4, 32);
        sm += __shfl_xor(sm, 8, 32);
        sm += __shfl_xor(sm, 16, 32);
        float attn = ex / sm;

        float* ag = agg + (size_t)d * DM + c0;
        atomicAdd(ag + 0, attn * vs.x);
        atomicAdd(ag + 1, attn * vs.y);
        atomicAdd(ag + 2, attn * vs.z);
        atomicAdd(ag + 3, attn * vs.w);
    }
}

// ---------------------------------------------------------------------------
// Fused residual + LayerNorm: out = LN(xin + res) * g + b. One wave per row.
// ---------------------------------------------------------------------------
__global__ void ln_add_kernel(const float* __restrict__ xin, const float* __restrict__ res,
                              const float* __restrict__ g, const float* __restrict__ b,
                              float* __restrict__ xout, int nrows) {
    int tid = threadIdx.x;
    int lane = tid & 31;
    int row = blockIdx.x * (blockDim.x >> 5) + (tid >> 5);
    if (row >= nrows) return;
    int c0 = lane * 4;
    const float4 xi = *(const float4*)(xin + (size_t)row * DM + c0);
    const float4 rr = *(const float4*)(res + (size_t)row * DM + c0);
    float y0 = xi.x + rr.x, y1 = xi.y + rr.y, y2 = xi.z + rr.z, y3 = xi.w + rr.w;
    float s = y0 + y1 + y2 + y3;
    for (int off = 1; off < 32; off <<= 1) s += __shfl_xor(s, off, 32);
    float mu = s * (1.0f / DM);
    float d0 = y0 - mu, d1 = y1 - mu, d2 = y2 - mu, d3 = y3 - mu;
    float vs = d0 * d0 + d1 * d1 + d2 * d2 + d3 * d3;
    for (int off = 1; off < 32; off <<= 1) vs += __shfl_xor(vs, off, 32);
    float inv = rsqrtf(vs * (1.0f / DM) + EPS);
    const float4 gg = *(const float4*)(g + c0);
    const float4 bb = *(const float4*)(b + c0);
    float4 o;
    o.x = d0 * inv * gg.x + bb.x;
    o.y = d1 * inv * gg.y + bb.y;
    o.z = d2 * inv * gg.z + bb.z;
    o.w = d3 * inv * gg.w + bb.w;
    *(float4*)(xout + (size_t)row * DM + c0) = o;
}

__global__ void zero_kernel(float* __restrict__ p, int n) {
    int i = blockIdx.x * blockDim.x + threadIdx.x;
    if (i < n) p[i] = 0.f;
}

// ---------------------------------------------------------------------------
extern "C" void kernel_launch(void* const* d_in, const int* in_sizes, int n_in,
                              void* d_out, int out_size, void* d_ws, size_t ws_size,
                              hipStream_t stream) {
    const float* nf    = (const float*)d_in[0];   // [N,20]
    const float* eattr = (const float*)d_in[1];   // [E,3]
    const int*   eidx  = (const int*)  d_in[2];   // [2,E]
    const float* embW  = (const float*)d_in[3];   // [20,128]
    const float* embB  = (const float*)d_in[4];   // [128]
    const float* Wq    = (const float*)d_in[5];   // [L,128,128]
    const float* Wk    = (const float*)d_in[6];
    const float* Wv    = (const float*)d_in[7];
    const float* eW1   = (const float*)d_in[8];   // [L,3,16]
    const float* eb1   = (const float*)d_in[9];   // [L,16]
    const float* eW2   = (const float*)d_in[10];  // [L,16,128]
    const float* eb2   = (const float*)d_in[11];  // [L,128]
    const float* lng   = (const float*)d_in[12];  // [L,128]
    const float* lnb   = (const float*)d_in[13];  // [L,128]
    const float* fW1   = (const float*)d_in[14];  // [L,128,512]
    const float* fb1   = (const float*)d_in[15];  // [L,512]
    const float* fW2   = (const float*)d_in[16];  // [L,512,128]
    const float* fb2   = (const float*)d_in[17];  // [L,128]

    const int N = in_sizes[0] / 20;
    const int E = in_sizes[1] / 3;
    const int* src = eidx;        // edge_index[0]
    const int* dst = eidx + E;    // edge_index[1]

    size_t nd = (size_t)N * DM;
    float* x    = (float*)d_ws;
    float* q    = x    + nd;
    float* kb   = q    + nd;
    float* vb   = kb   + nd;
    float* agg  = vb   + nd;
    float* ffo  = agg  + nd;
    float* hbuf = ffo  + nd;      // [N, FFDIM]

    dim3 blk(256);
    int rowTiles = (N + 15) / 16;       // N = 50000 -> exactly 3125 tiles
    int ndInt = (int)nd;

    embed_kernel<<<(N * DM + 255) / 256, blk, 0, stream>>>(nf, embW, embB, x, N);

    for (int l = 0; l < LAYERS; ++l) {
        gemm_wmma_f32<<<dim3(rowTiles, 1), blk, 0, stream>>>(
            x, Wq + (size_t)l * DM * DM, nullptr, q, N, DM, DM, 0);
        gemm_wmma_f32<<<dim3(rowTiles, 1), blk, 0, stream>>>(
            x, Wk + (size_t)l * DM * DM, nullptr, kb, N, DM, DM, 0);
        gemm_wmma_f32<<<dim3(rowTiles, 1), blk, 0, stream>>>(
            x, Wv + (size_t)l * DM * DM, nullptr, vb, N, DM, DM, 0);

        zero_kernel<<<(ndInt + 255) / 256, blk, 0, stream>>>(agg, ndInt);

        edge_attn_kernel<<<4096, blk, 0, stream>>>(
            q, kb, vb, eattr, src, dst,
            eW1 + (size_t)l * 48, eb1 + (size_t)l * 16,
            eW2 + (size_t)l * 16 * DM, eb2 + (size_t)l * DM, agg, E);

        ln_add_kernel<<<(N + 7) / 8, blk, 0, stream>>>(
            x, agg, lng + (size_t)l * DM, lnb + (size_t)l * DM, x, N);

        gemm_wmma_f32<<<dim3(rowTiles, 4), blk, 0, stream>>>(
            x, fW1 + (size_t)l * DM * FFDIM, fb1 + (size_t)l * FFDIM, hbuf, N, DM, FFDIM, 1);
        gemm_wmma_f32<<<dim3(rowTiles, 1), blk, 0, stream>>>(
            hbuf, fW2 + (size_t)l * FFDIM * DM, fb2 + (size_t)l * DM, ffo, N, FFDIM, DM, 0);

        float* outp = (l == LAYERS - 1) ? (float*)d_out : x;
        ln_add_kernel<<<(N + 7) / 8, blk, 0, stream>>>(
            x, ffo, lng + (size_t)l * DM, lnb + (size_t)l * DM, outp, N);
    }
}